// ConditionedConv1D_37194416783575
// MI455X (gfx1250) — compile-verified
//
#include <hip/hip_runtime.h>

// ---------------- problem constants (from reference) ----------------
#define B_      16
#define CIN     16
#define COUT    16
#define KSZ     3
#define DIL     2
#define CONDN   3
#define T_IN    262144
#define T_OUT   (T_IN - DIL * (KSZ - 1))   // 262140
#define JTOT    (KSZ * CIN)                 // 48 : reduction length
#define NCHUNK  (JTOT / 4)                  // 12 : K=4 WMMA steps
#define TILE_T  16
#define TILES_PER_WAVE 8
#define WAVE_T  (TILES_PER_WAVE * TILE_T)   // 128 t-positions per wave
#define STRIPS_PER_BATCH ((T_OUT + WAVE_T - 1) / WAVE_T)  // 2048 (exact power of 2)

typedef __attribute__((ext_vector_type(2))) float v2f;
typedef __attribute__((ext_vector_type(8))) float v8f;

// ---------------------------------------------------------------
// Phase 1: hypernetwork. kmatP[b][co][j'] with j' = k*16 + ci
// (reduction axis pre-permuted so WMMA A-fragments are contiguous
//  float2 per lane and each K-chunk shares a single tap k).
// ---------------------------------------------------------------
__global__ void hyper_kernel(const float* __restrict__ c,
                             const float* __restrict__ W,
                             const float* __restrict__ bias,
                             float* __restrict__ kmatP) {
    int idx = blockIdx.x * blockDim.x + threadIdx.x;
    if (idx >= B_ * COUT * JTOT) return;
    int bb = idx / (COUT * JTOT);
    int r  = idx % (COUT * JTOT);
    int co = r / JTOT;
    int jp = r % JTOT;          // j' = k*16 + ci
    int k  = jp >> 4;
    int ci = jp & 15;
    int f  = (co * CIN + ci) * KSZ + k;   // original flat row of W / bias
    float v = bias[f];
#pragma unroll
    for (int q = 0; q < CONDN; ++q)
        v += c[bb * CONDN + q] * W[f * CONDN + q];
    kmatP[idx] = v;
}

// ---------------------------------------------------------------
// Phase 2: per-batch GEMM  Y[16, T_out] = K[16,48] * Xpatch[48, T_out]
// via V_WMMA_F32_16X16X4_F32. One wave per 16(Cout) x 128(t) strip.
// ---------------------------------------------------------------
__global__ void __launch_bounds__(256)
conv_wmma_kernel(const float* __restrict__ x,
                 const float* __restrict__ kmatP,
                 float* __restrict__ y) {
    const int lane  = threadIdx.x & 31;
    const int wib   = threadIdx.x >> 5;               // wave in block: 0..7
    const int gwave = blockIdx.x * 8 + wib;           // global wave id
    const int bb    = gwave / STRIPS_PER_BATCH;       // batch (grid sized exactly)
    const int strip = gwave % STRIPS_PER_BATCH;

    const float* __restrict__ xb = x     + (size_t)bb * CIN  * T_IN;
    float*       __restrict__ yb = y     + (size_t)bb * COUT * T_OUT;
    const float* __restrict__ kb = kmatP + (size_t)bb * COUT * JTOT;

    const int half = lane >> 4;       // 0: lanes 0-15, 1: lanes 16-31
    const int l16  = lane & 15;
    const int sel  = half * 2;        // A/B VGPR0 holds K-idx sel, VGPR1 sel+1

    // ---- A fragments: a[kc].{x,y} = kmatP[bb][M=l16][4*kc + sel + {0,1}] ----
    v2f a[NCHUNK];
#pragma unroll
    for (int kc = 0; kc < NCHUNK; ++kc) {
        const float* p = kb + l16 * JTOT + 4 * kc + sel;  // 8B aligned
        a[kc] = *(const v2f*)p;
    }

    const int t0base = strip * WAVE_T;

    for (int tile = 0; tile < TILES_PER_WAVE; ++tile) {
        const int t0 = t0base + tile * TILE_T;
        const int t  = t0 + l16;                          // t <= T_IN-1 always
        // clamped x-column per tap (only needed for the very last tile)
        int tk[KSZ];
        tk[0] = t;
        tk[1] = (t + 2 < T_IN) ? (t + 2) : (T_IN - 1);
        tk[2] = (t + 4 < T_IN) ? (t + 4) : (T_IN - 1);

        v8f acc = {0.f, 0.f, 0.f, 0.f, 0.f, 0.f, 0.f, 0.f};

#pragma unroll
        for (int kc = 0; kc < NCHUNK; ++kc) {
            const int k   = kc >> 2;                      // tap index
            const int ci0 = (kc & 3) * 4 + sel;           // base input channel
            const int tc  = tk[k];
            v2f bfrag;
            bfrag.x = xb[(size_t)(ci0    ) * T_IN + tc];  // row ci0   : 64B coalesced
            bfrag.y = xb[(size_t)(ci0 + 1) * T_IN + tc];  // row ci0+1 : 64B coalesced
            acc = __builtin_amdgcn_wmma_f32_16x16x4_f32(
                    /*neg_a=*/false, a[kc],
                    /*neg_b=*/false, bfrag,
                    /*c_mod=*/(short)0, acc,
                    /*reuse_a=*/false, /*reuse_b=*/false);
        }

        // ---- store D: VGPR v holds Y[co = v + half*8][t] ----
        if (t < T_OUT) {
#pragma unroll
            for (int v = 0; v < 8; ++v) {
                const int co = v + half * 8;
                __builtin_nontemporal_store(acc[v],
                        &yb[(size_t)co * T_OUT + t]);     // streaming output
            }
        }
    }
}

extern "C" void kernel_launch(void* const* d_in, const int* in_sizes, int n_in,
                              void* d_out, int out_size, void* d_ws, size_t ws_size,
                              hipStream_t stream) {
    const float* x    = (const float*)d_in[0];  // [16,16,262144]
    const float* c    = (const float*)d_in[1];  // [16,3]
    const float* W    = (const float*)d_in[2];  // [768,3]
    const float* bias = (const float*)d_in[3];  // [768]
    float*       y    = (float*)d_out;          // [16,16,262140]
    float*       kmatP = (float*)d_ws;          // 16*16*48 floats = 48 KB scratch

    const int kelems = B_ * COUT * JTOT;        // 12288
    hipLaunchKernelGGL(hyper_kernel, dim3((kelems + 255) / 256), dim3(256), 0, stream,
                       c, W, bias, kmatP);

    const int total_waves = B_ * STRIPS_PER_BATCH;   // 32768
    hipLaunchKernelGGL(conv_wmma_kernel, dim3(total_waves / 8), dim3(256), 0, stream,
                       x, kmatP, y);
}